// RGCNBlock_31164282699786
// MI455X (gfx1250) — compile-verified
//
#include <hip/hip_runtime.h>

#define N_NODES 50000
#define NUM_E   800000
#define CH      256
#define NREL    8
#define BN_EPS  1e-5f

typedef float v2f __attribute__((ext_vector_type(2)));
typedef float v8f __attribute__((ext_vector_type(8)));

// ---------------------------------------------------------------- utilities
__global__ void zero_f32_kernel(float* __restrict__ p, long n) {
  long i = (long)blockIdx.x * blockDim.x + threadIdx.x;
  if (i < n) p[i] = 0.0f;
}

__global__ void init_h_bias_kernel(float* __restrict__ h, const float* __restrict__ bias) {
  long i = (long)blockIdx.x * blockDim.x + threadIdx.x;   // exactly N*CH threads
  h[i] = bias[(int)(i & (CH - 1))];
}

// ------------------------------------------------- edge scatter (one wave/edge)
__global__ void __launch_bounds__(256)
scatter_rel_kernel(const float* __restrict__ x,
                   const int*   __restrict__ ei,   // [2,E] flat: src row then dst row
                   const float* __restrict__ ea,   // [E,5], col 4 = relation id
                   float* __restrict__ msg,        // [N,CH] sums
                   float* __restrict__ cnt,        // [N]
                   int rel) {
  int e    = (int)(((long)blockIdx.x * blockDim.x + threadIdx.x) >> 5);
  int lane = threadIdx.x & 31;
  if (e >= NUM_E) return;
  int r = (int)ea[(long)e * 5 + 4];
  if (r != rel) return;
  int src = ei[e];
  int dst = ei[NUM_E + e];
  const float4* xs = (const float4*)(x + (long)src * CH);
  float* md = msg + (long)dst * CH;
#pragma unroll
  for (int j = 0; j < 2; ++j) {               // 8 channels per lane
    float4 v = xs[lane * 2 + j];
    int c = (lane * 2 + j) * 4;
    atomicAdd(md + c + 0, v.x);
    atomicAdd(md + c + 1, v.y);
    atomicAdd(md + c + 2, v.z);
    atomicAdd(md + c + 3, v.w);
  }
  if (lane == 0) atomicAdd(cnt + dst, 1.0f);
}

// ------------------------------------- H += rowscale(A) @ B  (f32 WMMA 16x16x4)
// A: [N,256] row-major, B: [256,256] row-major (in x out), H: [N,256]
// cnt != nullptr -> per-row scale 1/max(cnt[row],1)  (mean aggregation)
__global__ void __launch_bounds__(256)
wmma_gemm_accum_kernel(const float* __restrict__ A,
                       const float* __restrict__ B,
                       const float* __restrict__ cnt,
                       float* __restrict__ H) {
  __shared__ float Bp[CH * 16];               // 16 KB column panel [k][col]
  int n0 = blockIdx.y * 16;
  for (int j = threadIdx.x; j < CH * 16; j += 256) {
    int k = j >> 4, col = j & 15;
    Bp[j] = B[(long)k * CH + n0 + col];
  }
  __syncthreads();

  int wave   = threadIdx.x >> 5;
  int lane   = threadIdx.x & 31;
  int tile_m = blockIdx.x * 8 + wave;         // wave-uniform guard -> EXEC all 1s
  if (tile_m >= N_NODES / 16) return;
  int m0   = tile_m * 16;
  int half = lane >> 4;                       // 0: K0/K1 lanes, 1: K2/K3 lanes
  int lid  = lane & 15;
  int m    = m0 + lid;

  float s = 1.0f;
  if (cnt) s = 1.0f / fmaxf(cnt[m], 1.0f);
  const float* Arow = A + (long)m * CH;

  v8f c = {};
#pragma unroll 8
  for (int k = 0; k < CH; k += 4) {
    v2f a, b;
    a.x = Arow[k + 2 * half]     * s;
    a.y = Arow[k + 2 * half + 1] * s;
    b.x = Bp[(k + 2 * half) * 16 + lid];
    b.y = Bp[(k + 2 * half + 1) * 16 + lid];
    c = __builtin_amdgcn_wmma_f32_16x16x4_f32(false, a, false, b,
                                              (short)0, c, false, false);
  }
#pragma unroll
  for (int i = 0; i < 8; ++i) {               // C/D layout: vgpr i -> row i (+8 for hi half)
    int row = m0 + i + 8 * half;
    int col = n0 + lid;
    H[(long)row * CH + col] += c[i];
  }
}

// ----------------------------------------------------------- BatchNorm stats
__global__ void __launch_bounds__(256)
bn_stats_kernel(const float* __restrict__ H, float* __restrict__ stats) {
  int ch = threadIdx.x;                       // one channel per thread, coalesced rows
  int r0 = blockIdx.x * 256;
  int r1 = r0 + 256; if (r1 > N_NODES) r1 = N_NODES;
  float s = 0.0f, sq = 0.0f;
  for (int r = r0; r < r1; ++r) {
    float v = H[(long)r * CH + ch];
    s += v; sq += v * v;
  }
  atomicAdd(&stats[ch], s);
  atomicAdd(&stats[CH + ch], sq);
}

// ------------------------------------------------ BN + PReLU + residual fuse
__global__ void __launch_bounds__(256)
bn_prelu_res_kernel(float* __restrict__ H, const float* __restrict__ x,
                    const float* __restrict__ stats,
                    const float* __restrict__ gamma,
                    const float* __restrict__ beta,
                    const float* __restrict__ prelu_a) {
  long i = (long)blockIdx.x * blockDim.x + threadIdx.x;   // exactly N*CH threads
  int ch = (int)(i & (CH - 1));
  const float invN = 1.0f / (float)N_NODES;
  float mu  = stats[ch] * invN;
  float var = stats[CH + ch] * invN - mu * mu;
  float a   = prelu_a[0];
  float h   = H[i];
  h = (h - mu) * rsqrtf(var + BN_EPS) * gamma[ch] + beta[ch];
  h = (h > 0.0f) ? h : a * h;
  H[i] = h + x[i];
}

// --------------------------------------------------------------------- driver
extern "C" void kernel_launch(void* const* d_in, const int* in_sizes, int n_in,
                              void* d_out, int out_size, void* d_ws, size_t ws_size,
                              hipStream_t stream) {
  (void)in_sizes; (void)n_in; (void)out_size; (void)ws_size;
  const float* x     = (const float*)d_in[0];
  const int*   ei    = (const int*)  d_in[1];
  const float* ea    = (const float*)d_in[2];
  const float* W     = (const float*)d_in[3];   // [R,256,256]
  const float* root  = (const float*)d_in[4];   // [256,256]
  const float* bias  = (const float*)d_in[5];
  const float* gamma = (const float*)d_in[6];
  const float* beta  = (const float*)d_in[7];
  const float* pa    = (const float*)d_in[8];

  float* H     = (float*)d_out;                          // [N,256] accumulator
  float* msg   = (float*)d_ws;                           // [N,256] per-relation sums
  float* cnt   = msg + (long)N_NODES * CH;               // [N]
  float* stats = cnt + N_NODES;                          // [2*CH]

  const long hElems = (long)N_NODES * CH;                // 12.8M, /256 exact
  dim3 ggrid((N_NODES / 16 + 7) / 8, 16);                // 391 x 16, 8 waves/block

  init_h_bias_kernel<<<(unsigned)(hElems / 256), 256, 0, stream>>>(H, bias);
  wmma_gemm_accum_kernel<<<ggrid, 256, 0, stream>>>(x, root, nullptr, H);

  for (int r = 0; r < NREL; ++r) {
    long zn = hElems + N_NODES;                          // msg + cnt contiguous
    zero_f32_kernel<<<(unsigned)((zn + 255) / 256), 256, 0, stream>>>(msg, zn);
    scatter_rel_kernel<<<(unsigned)(((long)NUM_E * 32 + 255) / 256), 256, 0, stream>>>(
        x, ei, ea, msg, cnt, r);
    wmma_gemm_accum_kernel<<<ggrid, 256, 0, stream>>>(msg, W + (long)r * CH * CH, cnt, H);
  }

  zero_f32_kernel<<<1, 2 * CH, 0, stream>>>(stats, 2 * CH);
  bn_stats_kernel<<<(N_NODES + 255) / 256, 256, 0, stream>>>(H, stats);
  bn_prelu_res_kernel<<<(unsigned)(hElems / 256), 256, 0, stream>>>(
      H, x, stats, gamma, beta, pa);
}